// Enblock_9689446220621
// MI455X (gfx1250) — compile-verified
//
#include <hip/hip_runtime.h>
#include <cstdint>
#include <cstddef>

typedef float v2f __attribute__((ext_vector_type(2)));
typedef float v8f __attribute__((ext_vector_type(8)));

static __device__ __forceinline__ void atomicAddF32(float* p, float v) {
  __hip_atomic_fetch_add(p, v, __ATOMIC_RELAXED, __HIP_MEMORY_SCOPE_AGENT);
}

// ---------------- utility ----------------
__global__ void zero_f32(float* __restrict__ p, long n) {
  long i = (long)blockIdx.x * blockDim.x + threadIdx.x;
  if (i < n) p[i] = 0.0f;
}

// deg[row[e]] += 1
__global__ void degree_kernel(const long long* __restrict__ row, float* __restrict__ deg, int E) {
  int e = blockIdx.x * blockDim.x + threadIdx.x;
  if (e < E) atomicAddF32(&deg[(int)row[e]], 1.0f);
}

__global__ void dis_kernel(const float* __restrict__ deg, float* __restrict__ dis, int N) {
  int i = blockIdx.x * blockDim.x + threadIdx.x;
  if (i < N) {
    float d = deg[i];
    dis[i] = (d > 0.0f) ? rsqrtf(d) : 0.0f;
  }
}

__global__ void norm_kernel(const long long* __restrict__ row, const long long* __restrict__ col,
                            const float* __restrict__ dis, float* __restrict__ nrm, int E) {
  int e = blockIdx.x * blockDim.x + threadIdx.x;
  if (e < E) nrm[e] = -dis[(int)row[e]] * dis[(int)col[e]];
}

// dst[b, row[e], c] += src[b, col[e], c] * (coef * norm[e]); one wave per edge (32 lanes = 32 channels)
__global__ void prop_scatter(const float* __restrict__ src, float* __restrict__ dst,
                             const long long* __restrict__ row, const long long* __restrict__ col,
                             const float* __restrict__ nrm, float coef, int E, int N) {
  long idx = (long)blockIdx.x * blockDim.x + threadIdx.x;
  if (idx >= (long)E * 32) return;
  int e = (int)(idx >> 5);
  int c = (int)(idx & 31);
  int r  = (int)row[e];
  int cc = (int)col[e];
  float s = nrm[e] * coef;
#pragma unroll
  for (int b = 0; b < 4; ++b) {
    float v = src[((size_t)b * N + cc) * 32 + c] * s;
    atomicAddF32(&dst[((size_t)b * N + r) * 32 + c], v);
  }
}

// y -= x  (elementwise)
__global__ void sub_kernel(float* __restrict__ y, const float* __restrict__ x, long n) {
  long i = (long)blockIdx.x * blockDim.x + threadIdx.x;
  if (i < n) y[i] -= x[i];
}

// ---------------- WMMA GEMM: acc[M,64] += Tx[M,32] @ W[32,64] (f32 exact) ----------------
// W staged in LDS in k-paired layout: wlds[p*128 + n*2 + q] = W[2p+q][n]
// so each B fragment (two consecutive k-rows at column n) is a single aligned ds_load_b64.
__launch_bounds__(256)
__global__ void gemm_acc_wmma(const float* __restrict__ Tx, const float* __restrict__ W,
                              float* __restrict__ acc, int Mrows) {
  __shared__ float wlds[32 * 64];
  for (int i = threadIdx.x; i < 32 * 64; i += blockDim.x) {
    int p   = i >> 7;         // k-pair index 0..15
    int rem = i & 127;
    int n   = rem >> 1;       // column 0..63
    int q   = rem & 1;        // which of the two k-rows
    wlds[i] = W[(size_t)(2 * p + q) * 64 + n];
  }
  __syncthreads();

  const int wave = threadIdx.x >> 5;
  const int lane = threadIdx.x & 31;
  const int tile = blockIdx.x * 8 + wave;     // one 16x64 tile per wave
  const int ntiles = Mrows >> 4;
  if (tile >= ntiles) return;                 // wave-uniform: EXEC stays all-1s for WMMA

  const int half = lane >> 4;                 // K sub-slice select (per A/B VGPR layout)
  const int lm   = lane & 15;
  const int row0 = tile * 16;

  // load C accumulators (16x16 f32 layout: vgpr r -> row r + 8*half, col lm)
  v8f c0, c1, c2, c3;
  const float* accr = acc + (size_t)row0 * 64;
#pragma unroll
  for (int r = 0; r < 8; ++r) {
    size_t off = (size_t)(r + 8 * half) * 64 + lm;
    c0[r] = accr[off];
    c1[r] = accr[off + 16];
    c2[r] = accr[off + 32];
    c3[r] = accr[off + 48];
  }

  // A fragment base: lane lm holds row (row0+lm), K slice starts at 2*half
  const float* ap = Tx + (size_t)(row0 + lm) * 32 + 2 * half;

#pragma unroll
  for (int kk = 0; kk < 32; kk += 4) {
    v2f a = *(const v2f*)(ap + kk);               // A[m][kk+2*half], A[m][kk+2*half+1]
    int p = (kk >> 1) + half;                     // k-pair index for this lane's B slice
    const float* bp = &wlds[p * 128];
    v2f b0 = *(const v2f*)(bp + (lm)      * 2);   // W[2p..2p+1][lm]
    v2f b1 = *(const v2f*)(bp + (16 + lm) * 2);
    v2f b2 = *(const v2f*)(bp + (32 + lm) * 2);
    v2f b3 = *(const v2f*)(bp + (48 + lm) * 2);
    c0 = __builtin_amdgcn_wmma_f32_16x16x4_f32(false, a, false, b0, (short)0, c0, false, false);
    c1 = __builtin_amdgcn_wmma_f32_16x16x4_f32(false, a, false, b1, (short)0, c1, false, false);
    c2 = __builtin_amdgcn_wmma_f32_16x16x4_f32(false, a, false, b2, (short)0, c2, false, false);
    c3 = __builtin_amdgcn_wmma_f32_16x16x4_f32(false, a, false, b3, (short)0, c3, false, false);
  }

  float* accw = acc + (size_t)row0 * 64;
#pragma unroll
  for (int r = 0; r < 8; ++r) {
    size_t off = (size_t)(r + 8 * half) * 64 + lm;
    accw[off]      = c0[r];
    accw[off + 16] = c1[r];
    accw[off + 32] = c2[r];
    accw[off + 48] = c3[r];
  }
}

// ---------------- fused bias + ELU + sparse pool scatter ----------------
__global__ void pool_kernel(const float* __restrict__ h, const long long* __restrict__ prow,
                            const long long* __restrict__ pcol, const float* __restrict__ pval,
                            const float* __restrict__ bias, float* __restrict__ out,
                            int P, int N, int Mp) {
  long idx = (long)blockIdx.x * blockDim.x + threadIdx.x;
  if (idx >= (long)P * 64) return;
  int p = (int)(idx >> 6);
  int c = (int)(idx & 63);
  int r  = (int)prow[p];
  int cc = (int)pcol[p];
  float v = pval[p];
  float bb = bias[c];
#pragma unroll
  for (int b = 0; b < 4; ++b) {
    float x = h[((size_t)b * N + cc) * 64 + c] + bb;
    x = (x > 0.0f) ? x : (__expf(x) - 1.0f);   // ELU(alpha=1)
    atomicAddF32(&out[((size_t)b * Mp + r) * 64 + c], x * v);
  }
}

// ---------------- host orchestration ----------------
static inline size_t align256(size_t s) { return (s + 255) & ~(size_t)255; }

extern "C" void kernel_launch(void* const* d_in, const int* in_sizes, int n_in,
                              void* d_out, int out_size, void* d_ws, size_t ws_size,
                              hipStream_t stream) {
  const float*      x    = (const float*)d_in[0];      // [B,N,32]
  const long long*  edge = (const long long*)d_in[1];  // [2,E]
  const long long*  prow = (const long long*)d_in[2];  // [P]
  const long long*  pcol = (const long long*)d_in[3];  // [P]
  const float*      pval = (const float*)d_in[4];      // [P]
  const float*      wgt  = (const float*)d_in[5];      // [K,32,64]
  const float*      bias = (const float*)d_in[6];      // [64]
  float*            out  = (float*)d_out;              // [B,Mp,64]

  const int B = 4, Cin = 32, Cout = 64;
  const int E  = in_sizes[1] / 2;
  const int N  = in_sizes[0] / (B * Cin);
  const int P  = in_sizes[4];
  const int K  = in_sizes[5] / (Cin * Cout);
  const int Mp = out_size / (B * Cout);
  const long Mrows = (long)B * N;                      // GEMM rows (divisible by 16 here)

  const long long* erow = edge;
  const long long* ecol = edge + E;

  char* ws = (char*)d_ws;
  float* deg = (float*)ws;  ws += align256((size_t)N * 4);
  float* dis = (float*)ws;  ws += align256((size_t)N * 4);
  float* nrm = (float*)ws;  ws += align256((size_t)E * 4);
  float* tbuf[3];
  for (int i = 0; i < 3; ++i) { tbuf[i] = (float*)ws; ws += align256((size_t)Mrows * Cin * 4); }
  float* acc = (float*)ws;  ws += align256((size_t)Mrows * Cout * 4);

  const int T = 256;
  auto blk = [&](long n) { return (unsigned)((n + T - 1) / T); };

  // degree / dis / norm
  zero_f32<<<blk(N), T, 0, stream>>>(deg, N);
  degree_kernel<<<blk(E), T, 0, stream>>>(erow, deg, E);
  dis_kernel<<<blk(N), T, 0, stream>>>(deg, dis, N);
  norm_kernel<<<blk(E), T, 0, stream>>>(erow, ecol, dis, nrm, E);

  // acc = 0; acc += Tx0 @ W0  (Tx0 = x)
  const unsigned gemmBlocks = (unsigned)(((Mrows >> 4) + 7) / 8);
  zero_f32<<<blk(Mrows * Cout), T, 0, stream>>>(acc, Mrows * Cout);
  gemm_acc_wmma<<<gemmBlocks, T, 0, stream>>>(x, wgt, acc, (int)Mrows);

  // Tx1 = prop(x); acc += Tx1 @ W1
  const long scatterN = (long)E * 32;
  zero_f32<<<blk(Mrows * Cin), T, 0, stream>>>(tbuf[0], Mrows * Cin);
  prop_scatter<<<blk(scatterN), T, 0, stream>>>(x, tbuf[0], erow, ecol, nrm, 1.0f, E, N);
  if (K > 1) gemm_acc_wmma<<<gemmBlocks, T, 0, stream>>>(tbuf[0], wgt + 1 * Cin * Cout, acc, (int)Mrows);

  // Tx_k = 2*prop(Tx_{k-1}) - Tx_{k-2}; acc += Tx_k @ Wk   (rotating 3 buffers)
  const float* prev = x;
  const float* cur  = tbuf[0];
  int bi = 1;
  for (int k = 2; k < K; ++k) {
    float* nxt = tbuf[bi]; bi = (bi + 1) % 3;
    zero_f32<<<blk(Mrows * Cin), T, 0, stream>>>(nxt, Mrows * Cin);
    prop_scatter<<<blk(scatterN), T, 0, stream>>>(cur, nxt, erow, ecol, nrm, 2.0f, E, N);
    sub_kernel<<<blk(Mrows * Cin), T, 0, stream>>>(nxt, prev, Mrows * Cin);
    gemm_acc_wmma<<<gemmBlocks, T, 0, stream>>>(nxt, wgt + (size_t)k * Cin * Cout, acc, (int)Mrows);
    prev = cur; cur = nxt;
  }

  // out = 0; out[b,prow,:] += elu(acc[b,pcol,:] + bias) * pval
  zero_f32<<<blk(out_size), T, 0, stream>>>(out, out_size);
  pool_kernel<<<blk((long)P * 64), T, 0, stream>>>(acc, prow, pcol, pval, bias, out, P, N, Mp);
}